// IdealScoreMachine_22703197127254
// MI455X (gfx1250) — compile-verified
//
#include <hip/hip_runtime.h>
#include <hip/hip_bf16.h>

#define TIMESTEPS 1000
#define BETA_MIN 1e-4f
#define BETA_MAX 0.02f

#define Bq 16          // batch (query rows)
#define Nimg 32768     // dataset rows
#define Dfeat 3072     // 3*32*32

typedef __attribute__((ext_vector_type(2))) float v2f;
typedef __attribute__((ext_vector_type(8))) float v8f;

// ---- workspace layout (float offsets) ----
// [0..7]   sched: 0=at 1=bt 2=at^2 3=1/(2 bt^2) 4=bt^2
// [8..23]  x2[16]
// [24..39] denom[16]
// [64 .. 64+16*32768)        logits -> overwritten by w
// [64+16*32768 .. +16*3072)  wimg accumulator
#define WS_SCHED 0
#define WS_X2    8
#define WS_DENOM 24
#define WS_LOGIT 64
#define WS_WIMG  (64 + Bq * Nimg)

// ---------------- kernel 1: noise schedule (1 thread) ----------------
__global__ void sched_kernel(const int* __restrict__ tptr, float* __restrict__ ws) {
    if (threadIdx.x != 0 || blockIdx.x != 0) return;
    const int t = *tptr;
    const float step = (BETA_MAX - BETA_MIN) / (float)(TIMESTEPS - 1);
    float ab = 1.0f;
    for (int i = 0; i <= t; ++i) {
        float beta = BETA_MIN + step * (float)i;
        ab *= (1.0f - beta);
    }
    float at  = sqrtf(ab);
    float bt2 = 1.0f - ab;
    float bt  = sqrtf(bt2);
    ws[WS_SCHED + 0] = at;
    ws[WS_SCHED + 1] = bt;
    ws[WS_SCHED + 2] = at * at;
    ws[WS_SCHED + 3] = 1.0f / (2.0f * bt2);
    ws[WS_SCHED + 4] = bt2;
}

// ---------------- kernel 2: x row norms (16 blocks x 256) ----------------
__global__ void x2_kernel(const float* __restrict__ x, float* __restrict__ ws) {
    __shared__ float red[256];
    const int b = blockIdx.x;
    const float* row = x + (size_t)b * Dfeat;
    float s = 0.f;
    for (int d = threadIdx.x; d < Dfeat; d += 256) {
        float v = row[d];
        s += v * v;
    }
    red[threadIdx.x] = s;
    __syncthreads();
    for (int off = 128; off > 0; off >>= 1) {
        if (threadIdx.x < off) red[threadIdx.x] += red[threadIdx.x + off];
        __syncthreads();
    }
    if (threadIdx.x == 0) ws[WS_X2 + b] = red[0];
}

// ------- kernel 3: logits via WMMA f32 (dots fused with ||img||^2) -------
// One wave per 16-row N-tile. 2048 tiles, 8 waves/block -> 256 blocks x 256.
__global__ __launch_bounds__(256)
void logits_kernel(const float* __restrict__ x, const float* __restrict__ images,
                   float* __restrict__ ws) {
    const int wave  = threadIdx.x >> 5;
    const int tile  = blockIdx.x * 8 + wave;      // 0..2047
    const int lane  = threadIdx.x & 31;
    const int half  = lane >> 4;                  // 0: K=0,1  1: K=2,3
    const int l16   = lane & 15;
    const int n0    = tile * 16;

    // A: x row m=l16, k pair at +2*half; B: images row n0+l16, same k pair
    const float* arow = x + (size_t)l16 * Dfeat + 2 * half;
    const float* brow = images + (size_t)(n0 + l16) * Dfeat + 2 * half;

    v8f c = {};
    float i2p = 0.f;
    for (int k0 = 0; k0 < Dfeat; k0 += 4) {
        v2f a = *(const v2f*)(arow + k0);
        v2f b = *(const v2f*)(brow + k0);
        i2p += b.x * b.x + b.y * b.y;
        c = __builtin_amdgcn_wmma_f32_16x16x4_f32(
                /*neg_a=*/false, a, /*neg_b=*/false, b,
                /*c_mod=*/(short)0, c, /*reuse_a=*/false, /*reuse_b=*/false);
    }
    // combine lane & lane^16 partial square-sums -> full ||img_n||^2 per lane
    float i2n = i2p + __shfl_xor(i2p, 16, 32);

    const float at      = ws[WS_SCHED + 0];
    const float at2     = ws[WS_SCHED + 2];
    const float inv2bt2 = ws[WS_SCHED + 3];

    float* logits = ws + WS_LOGIT;
    #pragma unroll
    for (int j = 0; j < 8; ++j) {
        const int m = j + 8 * half;                         // C/D row
        const float dot = c[j];
        const float x2m = ws[WS_X2 + m];
        // logit = -(x2 - 2 at dot + at^2 i2) / (2 bt^2)
        const float logit = (2.0f * at * dot - x2m - at2 * i2n) * inv2bt2;
        logits[(size_t)m * Nimg + n0 + l16] = logit;
    }
}

// ------- kernel 4: per-row softmax weights (16 blocks x 256) -------
__global__ void softmax_kernel(float* __restrict__ ws) {
    __shared__ float red[256];
    const int b = blockIdx.x;
    float* row = ws + WS_LOGIT + (size_t)b * Nimg;

    // pass 1: max
    float mx = -3.4e38f;
    for (int n = threadIdx.x; n < Nimg; n += 256) mx = fmaxf(mx, row[n]);
    red[threadIdx.x] = mx;
    __syncthreads();
    for (int off = 128; off > 0; off >>= 1) {
        if (threadIdx.x < off) red[threadIdx.x] = fmaxf(red[threadIdx.x], red[threadIdx.x + off]);
        __syncthreads();
    }
    const float m = red[0];
    __syncthreads();

    // pass 2: w = exp(l - m), in place; accumulate denom
    float s = 0.f;
    for (int n = threadIdx.x; n < Nimg; n += 256) {
        float w = __expf(row[n] - m);
        row[n] = w;
        s += w;
    }
    red[threadIdx.x] = s;
    __syncthreads();
    for (int off = 128; off > 0; off >>= 1) {
        if (threadIdx.x < off) red[threadIdx.x] += red[threadIdx.x + off];
        __syncthreads();
    }
    if (threadIdx.x == 0) ws[WS_DENOM + b] = red[0];
}

// ------- kernel 5: w_img = w @ images via WMMA f32, K split 16 ways -------
// 192 D-tiles x 16 K-chunks = 3072 waves; 8 waves/block -> 384 blocks x 256.
#define KSPLIT 16
#define KCHUNK (Nimg / KSPLIT)   // 2048
__global__ __launch_bounds__(256)
void wimg_kernel(const float* __restrict__ images, float* __restrict__ ws) {
    const int wid   = blockIdx.x * 8 + (threadIdx.x >> 5);  // 0..3071
    const int tile  = wid / KSPLIT;                         // D tile 0..191
    const int chunk = wid % KSPLIT;
    const int lane  = threadIdx.x & 31;
    const int half  = lane >> 4;
    const int l16   = lane & 15;
    const int dcol  = tile * 16;
    const int kbeg  = chunk * KCHUNK;

    const float* w    = ws + WS_LOGIT;   // [16][Nimg]
    float*       wimg = ws + WS_WIMG;    // [16][Dfeat]

    // A: w row m=l16, k pair at +2*half (contiguous -> v2f)
    const float* arow = w + (size_t)l16 * Nimg + 2 * half;
    // B: images[k][dcol + l16], two consecutive k rows per lane
    const float* bbase = images + (size_t)(kbeg + 2 * half) * Dfeat + dcol + l16;

    v8f c = {};
    for (int k = 0; k < KCHUNK; k += 4) {
        v2f a = *(const v2f*)(arow + kbeg + k);
        v2f b;
        b.x = bbase[(size_t)k * Dfeat];
        b.y = bbase[(size_t)(k + 1) * Dfeat];
        c = __builtin_amdgcn_wmma_f32_16x16x4_f32(
                false, a, false, b, (short)0, c, false, false);
    }
    #pragma unroll
    for (int j = 0; j < 8; ++j) {
        const int m = j + 8 * half;
        atomicAdd(&wimg[(size_t)m * Dfeat + dcol + l16], c[j]);
    }
}

// ------- kernel 6: score = -(denom*x - at*wimg) / (denom*bt^2) -------
__global__ void final_kernel(const float* __restrict__ x, const float* __restrict__ ws,
                             float* __restrict__ out) {
    const int idx = blockIdx.x * 256 + threadIdx.x;
    if (idx >= Bq * Dfeat) return;
    const int b = idx / Dfeat;
    const float at    = ws[WS_SCHED + 0];
    const float bt2   = ws[WS_SCHED + 4];
    const float denom = ws[WS_DENOM + b];
    const float wi    = ws[WS_WIMG + idx];
    const float num   = denom * x[idx] - at * wi;
    out[idx] = -(num / (denom * bt2));
}

extern "C" void kernel_launch(void* const* d_in, const int* in_sizes, int n_in,
                              void* d_out, int out_size, void* d_ws, size_t ws_size,
                              hipStream_t stream) {
    const float* x      = (const float*)d_in[0];
    const float* images = (const float*)d_in[1];
    const int*   t      = (const int*)d_in[2];
    float* out = (float*)d_out;
    float* ws  = (float*)d_ws;

    // zero the atomic accumulator region (capture-safe async memset)
    hipMemsetAsync(ws + WS_WIMG, 0, (size_t)Bq * Dfeat * sizeof(float), stream);

    sched_kernel<<<1, 32, 0, stream>>>(t, ws);
    x2_kernel<<<Bq, 256, 0, stream>>>(x, ws);
    logits_kernel<<<Nimg / 16 / 8, 256, 0, stream>>>(x, images, ws);
    softmax_kernel<<<Bq, 256, 0, stream>>>(ws);
    wimg_kernel<<<(Dfeat / 16) * KSPLIT / 8, 256, 0, stream>>>(images, ws);
    final_kernel<<<(Bq * Dfeat + 255) / 256, 256, 0, stream>>>(x, ws, out);
}